// SparseVectorQuantizer_58987080843751
// MI455X (gfx1250) — compile-verified
//
#include <hip/hip_runtime.h>

// ---------------------------------------------------------------------------
// VQ codebook search on MI455X (gfx1250), FP32 WMMA path.
//   d2[n,k] = |z_n|^2 - 2 z.e + |e_k|^2 ; argmin_k == argmin_k (0.5|e_k|^2 - z.e)
// Compute-bound: 68.7 GFLOP vs ~35MB HBM traffic -> V_WMMA_F32_16X16X4_F32
// (fp16 would denormal-underflow on |e|~1.2e-4 products; bf16 mantissa too short
// for argmin fidelity). Codebook chunks are double-buffered in LDS and staged
// with GLOBAL_LOAD_ASYNC_TO_LDS_B128 (builtin confirmed present; takes int4*).
// ---------------------------------------------------------------------------

#ifndef __has_builtin
#define __has_builtin(x) 0
#endif

typedef __attribute__((ext_vector_type(2))) float v2f;
typedef __attribute__((ext_vector_type(8))) float v8f;

#define N_VOX 65536
#define K_EMB 8192
#define D_DIM 64
#define CHUNK 64            // codebook rows per stage (16 KB); x2 buffers = 32 KB

#if __has_builtin(__builtin_amdgcn_global_load_async_to_lds_b128)
#define HAVE_ASYNC_LDS 1
// Diagnostic showed the builtin wants pointers to 'int __vector_size__(16)'.
typedef int v4i_vs __attribute__((vector_size(16)));
typedef __attribute__((address_space(1))) v4i_vs gv4i;   // global src
typedef __attribute__((address_space(3))) v4i_vs lv4i;   // LDS dst
#else
#define HAVE_ASYNC_LDS 0
#endif

__device__ __forceinline__ void wait_async_keep4() {
#if HAVE_ASYNC_LDS
#if __has_builtin(__builtin_amdgcn_s_wait_asynccnt)
    __builtin_amdgcn_s_wait_asynccnt(4);
#else
    asm volatile("s_wait_asynccnt 0x4" ::: "memory");
#endif
#endif
}
__device__ __forceinline__ void wait_async_all() {
#if HAVE_ASYNC_LDS
#if __has_builtin(__builtin_amdgcn_s_wait_asynccnt)
    __builtin_amdgcn_s_wait_asynccnt(0);
#else
    asm volatile("s_wait_asynccnt 0x0" ::: "memory");
#endif
#endif
}

// Stage one 64x64 fp32 chunk of the codebook into LDS (4 x b128 per thread).
__device__ __forceinline__ void stage_chunk(const float* __restrict__ src,
                                            float* __restrict__ ldsDst, int tid) {
    const float4* s = (const float4*)src;
    float4* d = (float4*)ldsDst;
#if HAVE_ASYNC_LDS
#pragma unroll
    for (int i = 0; i < 4; ++i) {
        __builtin_amdgcn_global_load_async_to_lds_b128(
            (gv4i*)(uintptr_t)(s + tid + i * 256),   // flat->AS1 (identity bits)
            (lv4i*)(uintptr_t)(d + tid + i * 256),   // flat->AS3 (low 32b = offset)
            0, 0);
    }
#else
#pragma unroll
    for (int i = 0; i < 4; ++i)
        d[tid + i * 256] = s[tid + i * 256];
#endif
}

// ---------------------------------------------------------------------------
// Kernel 1: hesq[k] = 0.5 * sum_d e[k,d]^2 ; also zero the two loss slots.
// ---------------------------------------------------------------------------
__global__ void vq_prep(const float* __restrict__ e,
                        float* __restrict__ hesq,
                        float* __restrict__ losses) {
    const int k = blockIdx.x * blockDim.x + threadIdx.x;
    if (k == 0) { losses[0] = 0.0f; losses[1] = 0.0f; }
    if (k < K_EMB) {
        const float4* row = (const float4*)(e + (size_t)k * D_DIM);
        float s = 0.0f;
#pragma unroll
        for (int i = 0; i < D_DIM / 4; ++i) {
            float4 v = row[i];
            s += v.x * v.x + v.y * v.y + v.z * v.z + v.w * v.w;
        }
        hesq[k] = 0.5f * s;
    }
}

// ---------------------------------------------------------------------------
// Kernel 2: fused GEMM + argmin. 256 threads = 8 waves; each wave owns a
// 16-row z slab (A operands resident in 32 VGPRs, reused for all 8192 codes).
// Codebook chunks double-buffered in LDS; next stage's async loads issued
// before waiting on the current stage (async loads complete in order per ISA,
// so s_wait_asynccnt 4 leaves exactly the just-issued stage in flight).
//
// FP32 WMMA operand layout (ISA 7.12.2):
//   A 16x4: lanes 0-15 = M, VGPR0 = K0 (lo half)/K2 (hi half), VGPR1 = K1/K3
//   B 4x16: lanes 0-15 = N, same K striping (mirror of A)
//   C 16x16: slot r -> row r (lanes 0-15) / r+8 (lanes 16-31), N = lane%16
// Both A (z rows) and B (e rows) contract along the fast axis -> identical
// float2 pattern: rowptr + 4*c + 2*(lane>=16).
// ---------------------------------------------------------------------------
__launch_bounds__(256, 2)
__global__ void vq_argmin(const float* __restrict__ z,
                          const float* __restrict__ e,
                          const float* __restrict__ hesq,
                          int* __restrict__ idx_out,
                          float* __restrict__ idxf_out) {
    __shared__ float elds[2][CHUNK * D_DIM];   // 2 x 16 KB

    const int tid  = threadIdx.x;
    const int wave = tid >> 5;
    const int lane = tid & 31;
    const int lq   = lane & 15;       // N position within 16-wide tile
    const int half = lane >> 4;       // K-striping half (A/B) & row half (C)
    const int rowBase = blockIdx.x * 128 + wave * 16;

    // --- A slab: 16 rows x 64 D, this lane's stripe, resident in registers ---
    v2f a[16];
    const float* zrow = z + (size_t)(rowBase + lq) * D_DIM + 2 * half;
#pragma unroll
    for (int c = 0; c < 16; ++c)
        a[c] = *(const v2f*)(zrow + 4 * c);

    float mval[8];
    int   midx[8];
#pragma unroll
    for (int r = 0; r < 8; ++r) { mval[r] = 3.4e38f; midx[r] = 0; }

    // prologue: stage chunk 0 into buffer 0
    stage_chunk(e, &elds[0][0], tid);

    int buf = 0;
    for (int kc = 0; kc < K_EMB; kc += CHUNK, buf ^= 1) {
        const bool hasNext = (kc + CHUNK) < K_EMB;
        // issue next stage before waiting on the current one (overlap DMA/compute)
        if (hasNext)
            stage_chunk(e + (size_t)(kc + CHUNK) * D_DIM, &elds[buf ^ 1][0], tid);
        if (hasNext) wait_async_keep4(); else wait_async_all();
        __syncthreads();   // all waves' stage of this buffer complete & visible

#pragma unroll
        for (int t = 0; t < 4; ++t) {
            const int kbase = kc + t * 16;
            v8f acc0 = {0.f, 0.f, 0.f, 0.f, 0.f, 0.f, 0.f, 0.f};
            v8f acc1 = {0.f, 0.f, 0.f, 0.f, 0.f, 0.f, 0.f, 0.f};
            const float* brow = &elds[buf][0] + (t * 16 + lq) * D_DIM + 2 * half;
            // two independent accumulation chains for matrix-pipe ILP
#pragma unroll
            for (int c = 0; c < 16; c += 2) {
                v2f b0 = *(const v2f*)(brow + 4 * c);
                v2f b1 = *(const v2f*)(brow + 4 * c + 4);
                acc0 = __builtin_amdgcn_wmma_f32_16x16x4_f32(
                    false, a[c],     false, b0, (short)0, acc0, false, false);
                acc1 = __builtin_amdgcn_wmma_f32_16x16x4_f32(
                    false, a[c + 1], false, b1, (short)0, acc1, false, false);
            }
            // score = 0.5*|e_k|^2 - z.e   (argmin-equivalent to d2)
            const int   n = kbase + lq;
            const float h = hesq[n];
#pragma unroll
            for (int r = 0; r < 8; ++r) {
                float s = h - (acc0[r] + acc1[r]);
                if (s < mval[r]) { mval[r] = s; midx[r] = n; }
            }
        }
        __syncthreads();   // buffer may be overwritten by the stage after next
    }

    // min/argmin across the 16 lanes of each half (wave32 xor shuffles)
#pragma unroll
    for (int r = 0; r < 8; ++r) {
#pragma unroll
        for (int off = 1; off < 16; off <<= 1) {
            float ov = __shfl_xor(mval[r], off, 32);
            int   oi = __shfl_xor(midx[r], off, 32);
            if (ov < mval[r] || (ov == mval[r] && oi < midx[r])) {
                mval[r] = ov; midx[r] = oi;
            }
        }
    }
    if (lq == 0) {
        const int row0 = rowBase + 8 * half;   // lane0 -> rows 0..7, lane16 -> 8..15
#pragma unroll
        for (int r = 0; r < 8; ++r) {
            idx_out[row0 + r]  = midx[r];
            idxf_out[row0 + r] = (float)midx[r];
        }
    }
}

// ---------------------------------------------------------------------------
// Kernel 3: quantized_st[n,:] = e[idx[n],:]  (numerically == straight-through),
// plus mean((z-q)^2) into both loss slots (vq_loss == commitment_loss).
// ---------------------------------------------------------------------------
__global__ void vq_gather_loss(const float* __restrict__ z,
                               const float* __restrict__ e,
                               const int* __restrict__ idx,
                               float* __restrict__ qout,
                               float* __restrict__ losses) {
    const int g   = blockIdx.x * 256 + threadIdx.x;
    const int row = g >> 6;
    const int d   = g & 63;
    const int k   = idx[row];
    const float q  = e[(size_t)k * D_DIM + d];
    const float zv = z[g];
    qout[g] = q;

    float p = (zv - q) * (zv - q);
#pragma unroll
    for (int off = 16; off; off >>= 1)
        p += __shfl_xor(p, off, 32);

    __shared__ float wsum[8];
    const int lane = threadIdx.x & 31, wv = threadIdx.x >> 5;
    if (lane == 0) wsum[wv] = p;
    __syncthreads();
    if (wv == 0) {
        float s = (lane < 8) ? wsum[lane] : 0.0f;
#pragma unroll
        for (int off = 4; off; off >>= 1)
            s += __shfl_xor(s, off, 32);
        if (lane == 0) {
            const float v = s * (1.0f / (float)(N_VOX * D_DIM));
            atomicAdd(&losses[0], v);
            atomicAdd(&losses[1], v);
        }
    }
}

// ---------------------------------------------------------------------------
// Launch. Outputs concatenated: quantized_st [N*D] | vq_loss | commit | idxf [N]
// ws layout: int idx[N] | float hesq[K]  (288 KB)
// ---------------------------------------------------------------------------
extern "C" void kernel_launch(void* const* d_in, const int* in_sizes, int n_in,
                              void* d_out, int out_size, void* d_ws, size_t ws_size,
                              hipStream_t stream) {
    const float* z = (const float*)d_in[0];
    const float* e = (const float*)d_in[1];

    float* out    = (float*)d_out;
    float* qout   = out;
    float* losses = out + (size_t)N_VOX * D_DIM;
    float* idxf   = losses + 2;

    int*   idxi = (int*)d_ws;
    float* hesq = (float*)d_ws + N_VOX;

    vq_prep<<<K_EMB / 256, 256, 0, stream>>>(e, hesq, losses);
    vq_argmin<<<N_VOX / 128, 256, 0, stream>>>(z, e, hesq, idxi, idxf);
    vq_gather_loss<<<(N_VOX * D_DIM) / 256, 256, 0, stream>>>(z, e, idxi, qout, losses);
}